// ModShiftModel_61521111547944
// MI455X (gfx1250) — compile-verified
//
#include <hip/hip_runtime.h>
#include <math.h>

typedef __attribute__((ext_vector_type(2))) float v2f;
typedef __attribute__((ext_vector_type(8))) float v8f;

#define NPTS 4096          // 64*64 points

// -------- Pass 1: per-point squared norms (inputs are [C=3, 4096] channel-major)
__global__ void norms_kernel(const float* __restrict__ X, const float* __restrict__ Y,
                             float* __restrict__ nX, float* __restrict__ nY) {
    int i = blockIdx.x * blockDim.x + threadIdx.x;
    if (i < NPTS) {
        float x0 = X[i], x1 = X[NPTS + i], x2 = X[2 * NPTS + i];
        nX[i] = x0 * x0 + x1 * x1 + x2 * x2;
        float y0 = Y[i], y1 = Y[NPTS + i], y2 = Y[2 * NPTS + i];
        nY[i] = y0 * y0 + y1 * y1 + y2 * y2;
    }
}

// -------- Pass 2: Gram tiles via V_WMMA_F32_16X16X4_F32, branchless epilogue.
// wave = iTile*16 + jGroup; each wave does 16 j-tiles (256 cols) of its 16-row stripe.
__global__ void pairloss_kernel(const float* __restrict__ X, const float* __restrict__ Y,
                                const float* __restrict__ nX, const float* __restrict__ nY,
                                float* __restrict__ partials) {
    const int gtid = blockIdx.x * blockDim.x + threadIdx.x;
    const int wave = gtid >> 5;
    const int lane = threadIdx.x & 31;
    const int m  = lane & 15;   // row within A-tile / col within C-tile
    const bool hi = (lane >> 4) != 0;  // lane half selects K pair (0,1) vs (2,3)

    const int iTile  = wave >> 4;
    const int jGroup = wave & 15;
    const int i0 = iTile * 16;

    // A fragments (rows i0..i0+15, K padded to 4 with zero column).
    // Load all channels unconditionally (in-range, L2-resident), select with cndmask.
    const int rowA = i0 + m;
    const float xa0 = X[rowA], xa1 = X[NPTS + rowA], xa2 = X[2 * NPTS + rowA];
    const float ya0 = Y[rowA], ya1 = Y[NPTS + rowA], ya2 = Y[2 * NPTS + rowA];
    v2f ax, ay;
    ax[0] = hi ? xa2 : xa0;
    ax[1] = hi ? 0.0f : xa1;
    ay[0] = hi ? ya2 : ya0;
    ay[1] = hi ? 0.0f : ya1;

    // Row norms needed by this thread's 8 accumulator elements
    float nrX[8], nrY[8];
#pragma unroll
    for (int r = 0; r < 8; ++r) {
        const int row = i0 + r + (hi ? 8 : 0);
        nrX[r] = nX[row];
        nrY[r] = nY[row];
    }

    float acc = 0.0f;
    for (int jt = 0; jt < 16; ++jt) {
        const int j0 = (jGroup * 16 + jt) * 16;
        const int colB = j0 + m;

        // B fragments (cols j0..j0+15); same striping as A, branchless
        const float xb0 = X[colB], xb1 = X[NPTS + colB], xb2 = X[2 * NPTS + colB];
        const float yb0 = Y[colB], yb1 = Y[NPTS + colB], yb2 = Y[2 * NPTS + colB];
        v2f bx, by;
        bx[0] = hi ? xb2 : xb0;
        bx[1] = hi ? 0.0f : xb1;
        by[0] = hi ? yb2 : yb0;
        by[1] = hi ? 0.0f : yb1;

        v8f cx = {};
        v8f cy = {};
        // D = A x B + 0 : 16x16 tile of the Gram matrix, f32 WMMA (EXEC all-ones)
        cx = __builtin_amdgcn_wmma_f32_16x16x4_f32(
            false, ax, false, bx, (short)0, cx, false, false);
        cy = __builtin_amdgcn_wmma_f32_16x16x4_f32(
            false, ay, false, by, (short)0, cy, false, false);

        const float ncx = nX[colB];
        const float ncy = nY[colB];
#pragma unroll
        for (int r = 0; r < 8; ++r) {
            // ||xi-xj||^2 = ni + nj - 2*Gij, clamped; sqrt(0)=0 matches the guard
            const float sqx = fmaxf(nrX[r] + ncx - 2.0f * cx[r], 0.0f);
            const float sqy = fmaxf(nrY[r] + ncy - 2.0f * cy[r], 0.0f);
            const float dxv = __builtin_amdgcn_sqrtf(sqx);   // single v_sqrt_f32
            const float dyv = __builtin_amdgcn_sqrtf(sqy);
            const float t = fminf(dxv, 2.0f) - 1.0f;          // 2*min(d/2,1)-1
            const float w = fmaxf(1.0f - 0.5f * dyv, 0.0f);   // max(1-d/2,0)
            acc += t * w;
        }
    }

    // deterministic in-wave reduce (wave32)
#pragma unroll
    for (int off = 16; off > 0; off >>= 1)
        acc += __shfl_xor(acc, off, 32);
    if (lane == 0)
        partials[wave] = acc;
}

// -------- Pass 3: fixed-order reduction of 4096 wave partials -> scalar
__global__ void reduce_kernel(const float* __restrict__ partials, float* __restrict__ out) {
    __shared__ float s[256];
    const int t = threadIdx.x;
    float v = 0.0f;
    for (int k = t; k < NPTS; k += 256)
        v += partials[k];
    s[t] = v;
    __syncthreads();
    for (int off = 128; off > 0; off >>= 1) {
        if (t < off) s[t] += s[t + off];
        __syncthreads();
    }
    if (t == 0)
        out[0] = s[0] * 0.25f;  // 0.5 (loss_i) * 0.5 (final)
}

extern "C" void kernel_launch(void* const* d_in, const int* in_sizes, int n_in,
                              void* d_out, int out_size, void* d_ws, size_t ws_size,
                              hipStream_t stream) {
    (void)in_sizes; (void)n_in; (void)out_size; (void)ws_size;
    const float* X = (const float*)d_in[0];   // mv_points [1,3,64,64] -> [3,4096]
    const float* Y = (const float*)d_in[1];   // points0
    float* out = (float*)d_out;

    float* nX = (float*)d_ws;                 // 4096 f32
    float* nY = nX + NPTS;                    // 4096 f32
    float* partials = nY + NPTS;              // 4096 f32 (one per wave)

    norms_kernel<<<NPTS / 256, 256, 0, stream>>>(X, Y, nX, nY);
    // 4096 waves * 32 lanes = 131072 threads = 512 blocks of 256 (8 wave32 / block)
    pairloss_kernel<<<512, 256, 0, stream>>>(X, Y, nX, nY, partials);
    reduce_kernel<<<1, 256, 0, stream>>>(partials, out);
}